// GlobalAttention_11020886081537
// MI455X (gfx1250) — compile-verified
//
#include <hip/hip_runtime.h>
#include <hip/hip_bf16.h>
#include <math.h>

typedef __attribute__((ext_vector_type(2))) float v2f;
typedef __attribute__((ext_vector_type(8))) float v8f;

#define DCH 128
#define LDS_STRIDE 130           // padded row stride (floats) to avoid bank conflicts
#define WAVES_PER_BLOCK 4
#define EPS 1e-16f

// ---------------------------------------------------------------------------
// Kernel 1: gate[i] = relu(x[i,:] @ Wg1 + bg1) @ wg2 + bg2   via f32 WMMA
// One wave handles a 16-row tile of x. 8 column-tiles x 32 K-steps = 256 WMMAs.
// ---------------------------------------------------------------------------
__global__ __launch_bounds__(128) void gate_kernel(
    const float* __restrict__ x, const float* __restrict__ Wg1,
    const float* __restrict__ bg1, const float* __restrict__ wg2,
    const float* __restrict__ bg2, float* __restrict__ gate, int N)
{
  __shared__ float ldsW[DCH * LDS_STRIDE];   // Wg1 transposed: lds[n*130 + k]

  const int tid = threadIdx.x;
  for (int idx = tid; idx < DCH * DCH; idx += 128) {
    int k = idx >> 7, n = idx & 127;               // coalesced global read
    ldsW[n * LDS_STRIDE + k] = Wg1[idx];
  }
  __syncthreads();

  const int lane  = tid & 31;
  const int wv    = tid >> 5;
  const int m0    = (blockIdx.x * WAVES_PER_BLOCK + wv) * 16;
  if (m0 >= N) return;

  const int lhalf = lane >> 4;        // 0: K%4 in {0,1}, rows 0-7 of C
  const int l15   = lane & 15;        // A: row index; B/C: column index
  const int koff  = lhalf * 2;

  int arow = m0 + l15; if (arow >= N) arow = N - 1;   // clamp tail loads
  const float* ap = x + (size_t)arow * DCH + koff;
  const int bcol = l15 * LDS_STRIDE + koff;

  v8f zero = {0.f, 0.f, 0.f, 0.f, 0.f, 0.f, 0.f, 0.f};
  v8f acc[8];
#pragma unroll
  for (int j = 0; j < 8; ++j) acc[j] = zero;

#pragma unroll 4
  for (int kk = 0; kk < 32; ++kk) {
    v2f a = *(const v2f*)(ap + kk * 4);             // A frag: 16x4 of x
    const int kb = bcol + kk * 4;
#pragma unroll
    for (int j = 0; j < 8; ++j) {
      v2f b = *(const v2f*)(&ldsW[j * 16 * LDS_STRIDE + kb]);  // B frag: 4x16 of Wg1
      acc[j] = __builtin_amdgcn_wmma_f32_16x16x4_f32(
          false, a, false, b, (short)0, acc[j], false, false);
    }
  }

  // h = relu(acc + bg1); partial dot with wg2 along this lane's column
  float part[8];
#pragma unroll
  for (int i = 0; i < 8; ++i) part[i] = 0.f;
#pragma unroll
  for (int j = 0; j < 8; ++j) {
    const int n   = j * 16 + l15;
    const float b1 = bg1[n];
    const float w2 = wg2[n];
#pragma unroll
    for (int i = 0; i < 8; ++i) {
      float h = acc[j][i] + b1;
      h = fmaxf(h, 0.f);
      part[i] += h * w2;
    }
  }
  // reduce across the 16 lanes of each half (rows 0-7 live in lanes 0-15,
  // rows 8-15 in lanes 16-31; xor<=8 never crosses halves)
#pragma unroll
  for (int i = 0; i < 8; ++i) {
    float t = part[i];
    t += __shfl_xor(t, 1, 32);
    t += __shfl_xor(t, 2, 32);
    t += __shfl_xor(t, 4, 32);
    t += __shfl_xor(t, 8, 32);
    part[i] = t;
  }
  if (l15 == 0) {
    const float b2   = bg2[0];
    const int rbase  = m0 + lhalf * 8;
#pragma unroll
    for (int i = 0; i < 8; ++i) {
      int r = rbase + i;
      if (r < N) gate[r] = part[i] + b2;
    }
  }
}

// ---------------------------------------------------------------------------
// Kernel 2: segment start offsets from the sorted batch vector (int64)
// seg_start[s] = first i with batch[i] >= s ; seg_start[S] = N
// ---------------------------------------------------------------------------
__global__ void segstart_kernel(const long long* __restrict__ batch,
                                int* __restrict__ seg_start, int N, int S)
{
  int i = blockIdx.x * blockDim.x + threadIdx.x;
  if (i >= N) return;
  long long b  = batch[i];
  long long bp = (i == 0) ? -1ll : batch[i - 1];
  for (long long s = bp + 1; s <= b; ++s) seg_start[s] = i;
  if (i == N - 1) {
    for (long long s = b + 1; s <= (long long)S; ++s) seg_start[s] = N;
  }
}

// ---------------------------------------------------------------------------
// Kernel 3: per-segment softmax pooling + tiny GEMV against Wn.
// out[s] = (sum_i w_i * x_i) @ Wn + (sum_i w_i) * bn
// One block (128 threads) per segment; thread t owns channel t.
// ---------------------------------------------------------------------------
__global__ __launch_bounds__(128) void pool_kernel(
    const float* __restrict__ x, const float* __restrict__ gate,
    const int* __restrict__ seg_start, const float* __restrict__ Wn,
    const float* __restrict__ bn, float* __restrict__ out, int N)
{
  __shared__ float red[128];
  __shared__ float lw[128];
  __shared__ float pooled[128];

  const int s   = blockIdx.x;
  const int t   = threadIdx.x;
  const int beg = seg_start[s];
  const int end = seg_start[s + 1];

  // --- segment max of gate ---
  float m = -3.402823466e38f;
  for (int i = beg + t; i < end; i += 128) m = fmaxf(m, gate[i]);
  red[t] = m; __syncthreads();
  for (int o = 64; o > 0; o >>= 1) {
    if (t < o) red[t] = fmaxf(red[t], red[t + o]);
    __syncthreads();
  }
  m = red[0]; __syncthreads();

  // --- denominator ---
  float sum = 0.f;
  for (int i = beg + t; i < end; i += 128) sum += __expf(gate[i] - m);
  red[t] = sum; __syncthreads();
  for (int o = 64; o > 0; o >>= 1) {
    if (t < o) red[t] += red[t + o];
    __syncthreads();
  }
  const float denom = red[0];
  const float inv   = 1.f / (denom + EPS);
  const float swsum = denom * inv;     // sum of softmax weights (0 if empty)
  __syncthreads();

  // --- weighted pooling of x rows, chunked through LDS ---
  float acc = 0.f;
  for (int base = beg; base < end; base += 128) {
    int i = base + t;
    lw[t] = (i < end) ? __expf(gate[i] - m) * inv : 0.f;
    __syncthreads();
    int cnt = end - base; if (cnt > 128) cnt = 128;
    for (int q = 0; q < cnt; ++q)
      acc += lw[q] * x[(size_t)(base + q) * DCH + t];   // coalesced across t
    __syncthreads();
  }
  pooled[t] = acc;
  __syncthreads();

  // --- out[s] = pooled @ Wn + swsum*bn  (coalesced reads of Wn rows) ---
  float o = 0.f;
  for (int k = 0; k < DCH; ++k) o += pooled[k] * Wn[k * DCH + t];
  out[(size_t)s * DCH + t] = o + swsum * bn[t];
}

// ---------------------------------------------------------------------------
extern "C" void kernel_launch(void* const* d_in, const int* in_sizes, int n_in,
                              void* d_out, int out_size, void* d_ws, size_t ws_size,
                              hipStream_t stream) {
  const float*     x     = (const float*)d_in[0];
  const long long* batch = (const long long*)d_in[1];
  const float*     Wg1   = (const float*)d_in[2];
  const float*     bg1   = (const float*)d_in[3];
  const float*     wg2   = (const float*)d_in[4];
  const float*     bg2   = (const float*)d_in[5];
  const float*     Wn    = (const float*)d_in[6];
  const float*     bn    = (const float*)d_in[7];
  float*           out   = (float*)d_out;

  const int N = in_sizes[1];          // node count (batch length)
  const int S = out_size / DCH;       // segment count

  float* gate = (float*)d_ws;                 // N floats
  int*   seg  = (int*)(gate + N);             // S+1 ints

  const int tiles   = (N + 15) / 16;
  const int blocks1 = (tiles + WAVES_PER_BLOCK - 1) / WAVES_PER_BLOCK;

  gate_kernel<<<blocks1, 128, 0, stream>>>(x, Wg1, bg1, wg2, bg2, gate, N);
  segstart_kernel<<<(N + 255) / 256, 256, 0, stream>>>(batch, seg, N, S);
  pool_kernel<<<S, 128, 0, stream>>>(x, gate, seg, Wn, bn, out, N);
}